// Bottleneck_refine_30081950941421
// MI455X (gfx1250) — compile-verified
//
#include <hip/hip_runtime.h>

// ---------------------------------------------------------------------------
// SAR bottleneck (grouped 1x1 -> 3x3 -> 1x1, per-patch, masked, +residual, relu)
// One workgroup per (batch-patch, group). Full f32 via V_WMMA_F32_16X16X4_F32.
// ---------------------------------------------------------------------------

typedef __attribute__((ext_vector_type(2))) float v2f;
typedef __attribute__((ext_vector_type(8))) float v8f;

#define NB   16
#define NC   1024
#define NH   56
#define NW   56
#define NG   4
#define NMS  7
#define CPG  256   // conv1 input channels per group (= residual slab)
#define MIDG 64    // mid channels per group
#define COG  256   // conv3 output channels per group

__device__ __forceinline__ v8f wmma_f32(v2f a, v2f b, v8f c) {
  // D = A(16x4) * B(4x16) + C(16x16), f32 matrix pipe
  return __builtin_amdgcn_wmma_f32_16x16x4_f32(
      /*neg_a=*/false, a, /*neg_b=*/false, b,
      /*c_mod=*/(short)0, c, /*reuse_a=*/false, /*reuse_b=*/false);
}

__global__ __launch_bounds__(128) void sar_bneck_kernel(
    const float* __restrict__ x, const float* __restrict__ mask,
    const float* __restrict__ w1, const float* __restrict__ w2,
    const float* __restrict__ w3, float* __restrict__ out) {
  // LDS: x slab 64KB + padded conv1 out 25.6KB + conv2 out 16KB = ~106KB
  __shared__ float xg[CPG * 64];        // [ch][pix]   pix = y*8+x
  __shared__ float o1[MIDG * 100];      // [ch][10x10] zero halo (per-patch pad)
  __shared__ float o2[MIDG * 64];       // [ch][pix]

  const int wg  = blockIdx.x;
  const int g   = wg & 3;
  const int pid = wg >> 2;
  const int b   = pid / (NMS * NMS);
  const int p   = pid % (NMS * NMS);
  const int py  = p / NMS, px = p % NMS;
  const int y0  = py * 8, x0 = px * 8;

  const int tid  = threadIdx.x;
  const int lane = tid & 31;
  const int wave = tid >> 5;
  const int h    = lane >> 4;   // WMMA half: reg r, half h -> K = 2h + r
  const int ln   = lane & 15;   // M (A) or N (B/D) index within tile
  const int n    = wave * 16 + ln;       // this wave's pixel (N) column
  const int nyy  = n >> 3, nxx = n & 7;
  const int ppad = (nyy + 1) * 10 + (nxx + 1);

  const float* xbase = x  + ((size_t)(b * NC + g * CPG) * NH + y0) * NW + x0;
  float*       obase = out + ((size_t)(b * NC + g * COG) * NH + y0) * NW + x0;

  // ---- stage residual / conv1-input slab into LDS -------------------------
  for (int i = tid; i < CPG * 64; i += 128) {
    const int ch = i >> 6, pix = i & 63;
    xg[i] = xbase[(size_t)ch * NH * NW + (pix >> 3) * NW + (pix & 7)];
  }
  for (int i = tid; i < MIDG * 100; i += 128) o1[i] = 0.f;  // zero halo

  int msel = (mask[((b * NG + g) * NMS + py) * NMS + px] > 0.f) ? 1 : 0;
  msel = __builtin_amdgcn_readfirstlane(msel);  // scalar branch -> EXEC all-1s

  __syncthreads();

  if (msel) {
    // ---- conv1: o1 = relu(W1g[64,256] @ xg[256,64]) -----------------------
    {
      const float* w1g = w1 + (size_t)(g * MIDG) * CPG;  // [64 oc][256 ic]
      v8f acc[4] = {};
      for (int k = 0; k < CPG; k += 4) {
        const int k0 = k + 2 * h;
        v2f bm;
        bm.x = xg[(k0    ) * 64 + n];
        bm.y = xg[(k0 + 1) * 64 + n];
#pragma unroll
        for (int mt = 0; mt < 4; ++mt) {
          const v2f a = *(const v2f*)(w1g + (size_t)(mt * 16 + ln) * CPG + k0);
          acc[mt] = wmma_f32(a, bm, acc[mt]);
        }
      }
#pragma unroll
      for (int mt = 0; mt < 4; ++mt)
#pragma unroll
        for (int r = 0; r < 8; ++r) {
          float v = acc[mt][r];
          o1[(mt * 16 + r + 8 * h) * 100 + ppad] = v > 0.f ? v : 0.f;
        }
    }
    __syncthreads();

    // ---- conv2: o2 = relu(sum_t W2g_t[64,64] @ shift_t(o1)[64,64]) --------
    {
      const float* w2g = w2 + (size_t)(g * MIDG) * MIDG * 9;  // [oc][ic][3x3]
      v8f acc[4] = {};
#pragma unroll
      for (int t = 0; t < 9; ++t) {
        const int dy = t / 3 - 1, dx = t % 3 - 1;
        const int sp = (nyy + 1 + dy) * 10 + (nxx + 1 + dx);  // halo is zero
        for (int k = 0; k < MIDG; k += 4) {
          const int k0 = k + 2 * h;
          v2f bm;
          bm.x = o1[(k0    ) * 100 + sp];
          bm.y = o1[(k0 + 1) * 100 + sp];
#pragma unroll
          for (int mt = 0; mt < 4; ++mt) {
            const size_t wi = ((size_t)(mt * 16 + ln) * MIDG + k0) * 9 + t;
            v2f a;
            a.x = w2g[wi];
            a.y = w2g[wi + 9];
            acc[mt] = wmma_f32(a, bm, acc[mt]);
          }
        }
      }
      __syncthreads();  // o1 reads done before (not needed for o2 writes below)
#pragma unroll
      for (int mt = 0; mt < 4; ++mt)
#pragma unroll
        for (int r = 0; r < 8; ++r) {
          float v = acc[mt][r];
          o2[(mt * 16 + r + 8 * h) * 64 + n] = v > 0.f ? v : 0.f;
        }
    }
    __syncthreads();

    // ---- conv3 + residual + relu + store: W3g[256,64] @ o2[64,64] ---------
    {
      const float* w3g = w3 + (size_t)(g * COG) * MIDG;  // [256 oc][64 ic]
      for (int mt = 0; mt < 16; ++mt) {
        v8f acc = {};
        for (int k = 0; k < MIDG; k += 4) {
          const int k0 = k + 2 * h;
          v2f bm;
          bm.x = o2[(k0    ) * 64 + n];
          bm.y = o2[(k0 + 1) * 64 + n];
          const v2f a = *(const v2f*)(w3g + (size_t)(mt * 16 + ln) * MIDG + k0);
          acc = wmma_f32(a, bm, acc);
        }
#pragma unroll
        for (int r = 0; r < 8; ++r) {
          const int ocl = mt * 16 + r + 8 * h;       // out ch within group
          float v = acc[r] + xg[ocl * 64 + n];       // residual
          obase[(size_t)ocl * NH * NW + nyy * NW + nxx] = v > 0.f ? v : 0.f;
        }
      }
    }
  } else {
    // ---- masked patch: conv chain contributes zero -> out = relu(x) -------
    for (int i = tid; i < COG * 64; i += 128) {
      const int ch = i >> 6, pix = i & 63;
      const float v = xg[i];
      obase[(size_t)ch * NH * NW + (pix >> 3) * NW + (pix & 7)] =
          v > 0.f ? v : 0.f;
    }
  }
}

extern "C" void kernel_launch(void* const* d_in, const int* in_sizes, int n_in,
                              void* d_out, int out_size, void* d_ws,
                              size_t ws_size, hipStream_t stream) {
  (void)in_sizes; (void)n_in; (void)d_ws; (void)ws_size; (void)out_size;
  const float* x    = (const float*)d_in[0];
  const float* mask = (const float*)d_in[1];
  const float* w1   = (const float*)d_in[2];
  const float* w2   = (const float*)d_in[3];
  const float* w3   = (const float*)d_in[4];
  float* out = (float*)d_out;

  const dim3 grid(NB * NMS * NMS * NG);  // 3136 blocks: one per (patch, group)
  hipLaunchKernelGGL(sar_bneck_kernel, grid, dim3(128), 0, stream,
                     x, mask, w1, w2, w3, out);
}